// MeterAnomalyGNN_21474836480455
// MI455X (gfx1250) — compile-verified
//
#include <hip/hip_runtime.h>
#include <hip/hip_bf16.h>

typedef __attribute__((ext_vector_type(2))) float v2f;
typedef __attribute__((ext_vector_type(8))) float v8f;

// ---------------------------------------------------------------------------
// Zero workspace (graph-capture-safe)
// ---------------------------------------------------------------------------
__global__ void gnn_zero_kernel(float4* __restrict__ p, long n4) {
    long i = (long)blockIdx.x * blockDim.x + threadIdx.x;
    long stride = (long)gridDim.x * blockDim.x;
    for (; i < n4; i += stride) p[i] = make_float4(0.f, 0.f, 0.f, 0.f);
}

// ---------------------------------------------------------------------------
// Edge convolution with fp32 WMMA.
//   hout[dst] += [ relu?(hprev[src]) (K_SRC) || eattr (3) ] @ W + b
// One wave owns one 16-edge tile and ALL four 16-column output tiles:
//  - A (gathered source features) is fetched once per k-step (vs 4x before)
//  - 4 independent WMMA accumulation chains pipeline the matrix unit
// K padded to KPAD (multiple of 4); W zero-padded in LDS.
// ---------------------------------------------------------------------------
template<int K_SRC, bool RELU_IN>
__global__ void edge_conv_wmma(const float* __restrict__ hprev,   // [N, K_SRC]
                               const int*   __restrict__ eidx,    // [2, E]
                               const float* __restrict__ eattr,   // [E, 3]
                               const float* __restrict__ W,       // [K_SRC+3, 64]
                               const float* __restrict__ bias,    // [64]
                               float*       __restrict__ hout,    // [N, 64] zeroed
                               int nEdges) {
    constexpr int KIN  = K_SRC + 3;
    constexpr int KPAD = (KIN + 3) & ~3;
    __shared__ float Wlds[KPAD * 64];

    for (int i = threadIdx.x; i < KPAD * 64; i += blockDim.x) {
        int r = i >> 6;
        Wlds[i] = (r < KIN) ? W[i] : 0.0f;
    }
    __syncthreads();

    const int lane = threadIdx.x & 31;
    const int wave = threadIdx.x >> 5;
    const int stream   = blockIdx.x * 8 + wave;     // 8 waves/block, 1 tile stream each
    const int nStreams = gridDim.x * 8;
    const int nTiles   = nEdges >> 4;               // E divisible by 16
    const int row  = lane & 15;                     // edge row in tile / N index
    const int hi   = lane >> 4;                     // lane half: K +2, M +8

    float bvec[4];
#pragma unroll
    for (int ct = 0; ct < 4; ++ct) bvec[ct] = bias[ct * 16 + row];

    for (int t = stream; t < nTiles; t += nStreams) {
        const int base  = t << 4;
        const int s_row = eidx[base + row];             // src of my row
        const int d_row = eidx[nEdges + base + row];    // dst of my row
        const float ea0 = eattr[(base + row) * 3 + 0];
        const float ea1 = eattr[(base + row) * 3 + 1];
        const float ea2 = eattr[(base + row) * 3 + 2];

        v8f c0 = {}, c1 = {}, c2 = {}, c3 = {};
#pragma unroll
        for (int k0 = 0; k0 < KPAD; k0 += 4) {
            const int ka = k0 + (hi << 1);
            float a0, a1;
            if (ka < K_SRC) {
                float v = hprev[(long)s_row * K_SRC + ka];
                a0 = RELU_IN ? fmaxf(v, 0.f) : v;
            } else if (ka < KIN) {
                a0 = (ka == K_SRC) ? ea0 : ((ka == K_SRC + 1) ? ea1 : ea2);
            } else {
                a0 = 0.f;
            }
            const int kb = ka + 1;
            if (kb < K_SRC) {
                float v = hprev[(long)s_row * K_SRC + kb];
                a1 = RELU_IN ? fmaxf(v, 0.f) : v;
            } else if (kb < KIN) {
                a1 = (kb == K_SRC) ? ea0 : ((kb == K_SRC + 1) ? ea1 : ea2);
            } else {
                a1 = 0.f;
            }
            v2f a; a.x = a0; a.y = a1;

            const float* wr0 = &Wlds[(k0 + (hi << 1) + 0) * 64 + row];
            const float* wr1 = &Wlds[(k0 + (hi << 1) + 1) * 64 + row];
            v2f b;
            b.x = wr0[0];  b.y = wr1[0];
            c0 = __builtin_amdgcn_wmma_f32_16x16x4_f32(false, a, false, b, (short)0, c0, false, false);
            b.x = wr0[16]; b.y = wr1[16];
            c1 = __builtin_amdgcn_wmma_f32_16x16x4_f32(false, a, false, b, (short)0, c1, false, false);
            b.x = wr0[32]; b.y = wr1[32];
            c2 = __builtin_amdgcn_wmma_f32_16x16x4_f32(false, a, false, b, (short)0, c2, false, false);
            b.x = wr0[48]; b.y = wr1[48];
            c3 = __builtin_amdgcn_wmma_f32_16x16x4_f32(false, a, false, b, (short)0, c3, false, false);
        }

        // Scatter-add: lane half 'hi' holds rows M = hi*8 + v.
#pragma unroll
        for (int v = 0; v < 8; ++v) {
            int m = v + (hi << 3);
            int d = __shfl(d_row, m, 32);
            float* o = &hout[(long)d * 64 + row];
            atomicAdd(o +  0, c0[v] + bvec[0]);
            atomicAdd(o + 16, c1[v] + bvec[1]);
            atomicAdd(o + 32, c2[v] + bvec[2]);
            atomicAdd(o + 48, c3[v] + bvec[3]);
        }
    }
}

// ---------------------------------------------------------------------------
// out[n] = relu(h2[n]) . W3 + b3   (one node per wave, wave32 reduce)
// ---------------------------------------------------------------------------
__global__ void gnn_out_kernel(const float* __restrict__ h2,
                               const float* __restrict__ W3,
                               const float* __restrict__ b3,
                               float* __restrict__ out, int nNodes) {
    int node = blockIdx.x * (blockDim.x >> 5) + (threadIdx.x >> 5);
    int lane = threadIdx.x & 31;
    if (node >= nNodes) return;
    float w0 = W3[lane], w1 = W3[lane + 32];
    const float* hrow = h2 + (long)node * 64;
    float p = fmaxf(hrow[lane], 0.f) * w0 + fmaxf(hrow[lane + 32], 0.f) * w1;
#pragma unroll
    for (int off = 16; off; off >>= 1) p += __shfl_down(p, off, 32);
    if (lane == 0) out[node] = p + b3[0];
}

// ---------------------------------------------------------------------------
extern "C" void kernel_launch(void* const* d_in, const int* in_sizes, int n_in,
                              void* d_out, int out_size, void* d_ws, size_t ws_size,
                              hipStream_t stream) {
    const float* x     = (const float*)d_in[0];
    const int*   eidx  = (const int*)  d_in[1];
    const float* eattr = (const float*)d_in[2];
    const float* W1    = (const float*)d_in[3];
    const float* b1    = (const float*)d_in[4];
    const float* W2    = (const float*)d_in[5];
    const float* b2    = (const float*)d_in[6];
    const float* W3    = (const float*)d_in[7];
    const float* b3    = (const float*)d_in[8];
    float* out = (float*)d_out;

    const int nNodes = in_sizes[0] / 3;   // 100000
    const int nEdges = in_sizes[1] / 2;   // 1600000

    float* h1 = (float*)d_ws;                       // [N,64]
    float* h2 = h1 + (size_t)nNodes * 64;           // [N,64]

    const long n4 = ((long)nNodes * 64 * 2) / 4;
    gnn_zero_kernel<<<2048, 256, 0, stream>>>((float4*)d_ws, n4);

    // Layer 1: K_SRC=3 (KPAD=8), no input relu.
    edge_conv_wmma<3, false><<<1024, 256, 0, stream>>>(
        x, eidx, eattr, W1, b1, h1, nEdges);

    // Layer 2: K_SRC=64 (KPAD=68), relu on gathered h1.
    edge_conv_wmma<64, true><<<2048, 256, 0, stream>>>(
        h1, eidx, eattr, W2, b2, h2, nEdges);

    // Output projection.
    gnn_out_kernel<<<(nNodes + 7) / 8, 256, 0, stream>>>(h2, W3, b3, out, nNodes);
}